// Loss_for_localization_55551107006911
// MI455X (gfx1250) — compile-verified
//
#include <hip/hip_runtime.h>
#include <hip/hip_bf16.h>

// ---------------------------------------------------------------------------
// Problem constants (match reference)
// ---------------------------------------------------------------------------
static constexpr int BB = 128;           // batch
static constexpr int HH = 224;
static constexpr int WW = 224;
static constexpr int HWp = HH * WW;      // 50176 floats per plane
static constexpr int HW4 = HWp / 4;      // 12544 float4 chunks per plane
static constexpr int NCLS = 1000;

static constexpr int NBLK = 1024;        // blocks for the streaming kernel
static constexpr int NTHR = 256;         // 8 wave32 per block

// ws layout (floats):
//   [0      , 1024)  per-block noobj partials
//   [1024   , 2048)  per-block obj partials
//   [2048   , 3072)  per-block coor partials
//   [3072   , 3200)  per-batch-row class NLL (128 entries)
static constexpr int WS_NO  = 0;
static constexpr int WS_OB  = 1024;
static constexpr int WS_CO  = 2048;
static constexpr int WS_CLS = 3072;

typedef float v4f __attribute__((ext_vector_type(4)));
typedef float v2f __attribute__((ext_vector_type(2)));
typedef float v8f __attribute__((ext_vector_type(8)));

// ---------------------------------------------------------------------------
// Kernel 1: stream objects/gt/locs, accumulate the three masked sums.
// Memory-bound: 10 x b128 NT loads per float4 chunk; everything coalesced.
// ---------------------------------------------------------------------------
__global__ __launch_bounds__(NTHR) void pixel_loss_kernel(
    const float* __restrict__ objects,   // (B,H,W)
    const float* __restrict__ locs,      // (B,4,H,W)
    const float* __restrict__ gt,        // (B,5,H,W)
    float* __restrict__ ws) {
  float accN = 0.0f, accO = 0.0f, accC = 0.0f;

  const int total = BB * HW4;
  for (int t = blockIdx.x * NTHR + threadIdx.x; t < total;
       t += gridDim.x * NTHR) {
    const int b = t / HW4;
    const int r = (t - b * HW4) * 4;          // float offset within plane

    const float* obase = objects + (size_t)b * HWp + r;
    const float* gbase = gt      + (size_t)b * 5 * HWp + r;
    const float* lbase = locs    + (size_t)b * 4 * HWp + r;

    // single-use 257MB stream > 192MB L2 -> non-temporal loads
    v4f p  = __builtin_nontemporal_load((const v4f*)obase);
    v4f g0 = __builtin_nontemporal_load((const v4f*)gbase);

    v4f sq = {0.0f, 0.0f, 0.0f, 0.0f};
#pragma unroll
    for (int c = 0; c < 4; ++c) {
      v4f l = __builtin_nontemporal_load((const v4f*)(lbase + (size_t)c * HWp));
      v4f g = __builtin_nontemporal_load((const v4f*)(gbase + (size_t)(c + 1) * HWp));
      v4f d = l - g;
      sq += d * d;
    }

#pragma unroll
    for (int e = 0; e < 4; ++e) {
      const float ge = g0[e];
      const float pe = p[e];
      // branch-free: compute both transcendentals, select with cndmask
      const float ln1m = fmaxf(__logf(1.0f - pe), -100.0f); // log1p(-p), p in [eps,1-eps]
      const float lnp  = fmaxf(__logf(pe),        -100.0f);
      const bool isNo  = (ge == 0.0f);
      const bool isOb  = (ge == 1.0f);
      accN -= isNo ? ln1m  : 0.0f;
      accO -= isOb ? lnp   : 0.0f;
      accC += isOb ? sq[e] : 0.0f;
    }
  }

  // deterministic block reduction (no float atomics)
  __shared__ float sN[NTHR], sO[NTHR], sC[NTHR];
  const int tid = threadIdx.x;
  sN[tid] = accN; sO[tid] = accO; sC[tid] = accC;
  __syncthreads();
#pragma unroll
  for (int s = NTHR / 2; s > 0; s >>= 1) {
    if (tid < s) {
      sN[tid] += sN[tid + s];
      sO[tid] += sO[tid + s];
      sC[tid] += sC[tid + s];
    }
    __syncthreads();
  }
  if (tid == 0) {
    ws[WS_NO + blockIdx.x] = sN[0];
    ws[WS_OB + blockIdx.x] = sO[0];
    ws[WS_CO + blockIdx.x] = sC[0];
  }
}

// ---------------------------------------------------------------------------
// Kernel 2: per-row log-softmax NLL over 1000 classes. One block per row.
// ---------------------------------------------------------------------------
__global__ __launch_bounds__(NTHR) void class_loss_kernel(
    const float* __restrict__ scores,    // (B, 1000)
    const int* __restrict__ label,       // (B,)
    float* __restrict__ ws) {
  const int b = blockIdx.x;
  const float* row = scores + (size_t)b * NCLS;
  __shared__ float red[NTHR];
  const int t = threadIdx.x;

  float mx = -3.402823466e+38f;
  for (int i = t; i < NCLS; i += NTHR) mx = fmaxf(mx, row[i]);
  red[t] = mx;
  __syncthreads();
#pragma unroll
  for (int s = NTHR / 2; s > 0; s >>= 1) {
    if (t < s) red[t] = fmaxf(red[t], red[t + s]);
    __syncthreads();
  }
  mx = red[0];
  __syncthreads();

  float sum = 0.0f;
  for (int i = t; i < NCLS; i += NTHR) sum += __expf(row[i] - mx);
  red[t] = sum;
  __syncthreads();
#pragma unroll
  for (int s = NTHR / 2; s > 0; s >>= 1) {
    if (t < s) red[t] += red[t + s];
    __syncthreads();
  }
  if (t == 0) {
    const float logz = row[label[b]] - mx - __logf(red[0]);
    ws[WS_CLS + b] = -logz;   // per-row NLL
  }
}

// ---------------------------------------------------------------------------
// Kernel 3: WMMA-based final reduction + scalar combine. One wave32.
//
// V_WMMA_F32_16X16X4_F32 with A = ones(16x4):
//   D[m][n] = sum_k B[k][n] + C[m][n]
// so sum_n D[0][n] == sum of ALL 64 elements of B, independent of where each
// element lands in the B layout. We chain-accumulate C across 64-float chunks,
// then fold the 16 columns of row 0 with 4 xor-shuffles.
// ---------------------------------------------------------------------------
__device__ __forceinline__ float wmma_sum_region(const float* __restrict__ src,
                                                 int n /* multiple of 64 */) {
  const int lane = threadIdx.x;          // 0..31, wave32
  v2f a; a.x = 1.0f; a.y = 1.0f;         // A = ones (layout-invariant)
  v8f c = {};
  for (int base = 0; base < n; base += 64) {
    v2f bm;
    bm.x = src[base + lane];
    bm.y = src[base + 32 + lane];
    c = __builtin_amdgcn_wmma_f32_16x16x4_f32(
        /*neg_a=*/false, a, /*neg_b=*/false, bm,
        /*c_mod=*/(short)0, c, /*reuse_a=*/false, /*reuse_b=*/false);
  }
  // c[0]: lanes 0..15 = row0 col n; lanes 16..31 = row8 (identical).
  float v = c[0];
#pragma unroll
  for (int off = 8; off >= 1; off >>= 1)
    v += __shfl_xor(v, off, 16);
  return v;                               // valid in lanes 0..15
}

__global__ __launch_bounds__(32) void final_combine_kernel(
    const float* __restrict__ ws,
    const float* __restrict__ obj_coor,
    const float* __restrict__ no_obj_confi,
    const float* __restrict__ img_class_weight,
    float* __restrict__ out) {
  const float sNo  = wmma_sum_region(ws + WS_NO,  NBLK);
  const float sOb  = wmma_sum_region(ws + WS_OB,  NBLK);
  const float sCo  = wmma_sum_region(ws + WS_CO,  NBLK);
  const float sCls = wmma_sum_region(ws + WS_CLS, BB);   // 128 = 2 chunks
  if (threadIdx.x == 0) {
    const float img_class_loss = sCls / (float)BB;
    out[0] = img_class_weight[0] * img_class_loss +
             (no_obj_confi[0] * sNo + sOb + obj_coor[0] * sCo) / (float)BB;
  }
}

// ---------------------------------------------------------------------------
// Host launcher
// ---------------------------------------------------------------------------
extern "C" void kernel_launch(void* const* d_in, const int* in_sizes, int n_in,
                              void* d_out, int out_size, void* d_ws, size_t ws_size,
                              hipStream_t stream) {
  (void)in_sizes; (void)n_in; (void)out_size; (void)ws_size;
  const float* objects          = (const float*)d_in[0];
  const float* scores           = (const float*)d_in[1];
  const float* locs             = (const float*)d_in[2];
  const int*   label            = (const int*)d_in[3];
  const float* gt               = (const float*)d_in[4];
  const float* obj_coor         = (const float*)d_in[5];
  const float* no_obj_confi     = (const float*)d_in[6];
  const float* img_class_weight = (const float*)d_in[7];
  float* out = (float*)d_out;
  float* ws  = (float*)d_ws;

  pixel_loss_kernel<<<NBLK, NTHR, 0, stream>>>(objects, locs, gt, ws);
  class_loss_kernel<<<BB, NTHR, 0, stream>>>(scores, label, ws);
  final_combine_kernel<<<1, 32, 0, stream>>>(ws, obj_coor, no_obj_confi,
                                             img_class_weight, out);
}